// CRF_76647986364981
// MI455X (gfx1250) — compile-verified
//
#include <hip/hip_runtime.h>
#include <hip/hip_bf16.h>

#define S 512
#define B 1024
#define T 48
#define START 46
#define STOP 47
#define NEG -10000.0f

typedef float v2f __attribute__((ext_vector_type(2)));
typedef float v8f __attribute__((ext_vector_type(8)));

// Async copy of one 3KB feats tile (16 batches x 48 tags, contiguous) into LDS.
// 96 threads x 2 x b128 = 3072 B. Tracked on ASYNCcnt (2 instructions / wave).
__device__ __forceinline__ void async_load_tile(const float* __restrict__ gbase,
                                                unsigned ldsbase, int tid) {
    const unsigned long long sb = (unsigned long long)(uintptr_t)gbase;  // uniform
    unsigned voff = (unsigned)tid * 16u;
    unsigned la   = ldsbase + voff;
    asm volatile("global_load_async_to_lds_b128 %0, %1, %2 offset:0"
                 :: "v"(la), "v"(voff), "s"(sb) : "memory");
    asm volatile("global_load_async_to_lds_b128 %0, %1, %2 offset:0"
                 :: "v"(la + 1536u), "v"(voff + 1536u), "s"(sb) : "memory");
}

// ---------------------------------------------------------------------------
// Kernel 1: CRF forward algorithm.
// One block per 16-batch tile (64 blocks). 96 threads = 3 waves; wave w owns
// output-tag tile n in [16w, 16w+16). alpha (16x48) lives in LDS for all 512
// steps. Per step:
//   m[b] = max_j alpha[b,j];  P[b,j] = exp(alpha[b,j]-m[b])
//   D = P @ E^T via 12x V_WMMA_F32_16X16X4_F32 per wave (E = exp(trans),
//       constant B-fragments preloaded in 24 VGPRs)
//   alpha[b,i] = alive ? feat[t,b,i] + m[b] + log(D[b,i]) : alpha[b,i]
// feats tiles are double-buffered and streamed one step ahead with
// global_load_async_to_lds_b128 (ASYNCcnt), hiding HBM latency under the
// sequential recurrence.
// ---------------------------------------------------------------------------
__global__ void crf_forward_kernel(const float* __restrict__ feats,
                                   const float* __restrict__ trans,
                                   const int* __restrict__ lens,
                                   float* __restrict__ fwd) {
    __shared__ float alphaS[16 * 48];
    __shared__ float Ps[16 * 48];
    __shared__ __align__(16) float featS[2][16 * 48];
    __shared__ float mS[16];
    __shared__ float redS[6 * 16];
    __shared__ int   lensS[16];

    const int tid   = threadIdx.x;        // 0..95
    const int lane  = tid & 31;           // wave32
    const int wave  = tid >> 5;           // 0..2
    const int nbase = wave * 16;          // output-tag tile
    const int hi    = lane >> 4;          // 0 / 1 (lane half)
    const int nl    = lane & 15;          // N (and A's M) index within tile
    const int b0    = blockIdx.x * 16;    // batch tile base

    const unsigned lds0 = (unsigned)(uintptr_t)&featS[0][0];
    const unsigned lds1 = (unsigned)(uintptr_t)&featS[1][0];

    // ---- init alpha, lens ----
    for (int idx = tid; idx < 16 * 48; idx += 96)
        alphaS[idx] = ((idx % 48) == START) ? 0.0f : NEG;
    if (tid < 16) lensS[tid] = lens[b0 + tid];

    // ---- preload B fragments: B[k][n] = E^T[k][n] = exp(trans[nbase+n][k]).
    // Per-lane layout (mirrors C/D row striping): component c holds
    // K = 4*kc + 2*hi + c, N = lane%16.
    v2f bf[12];
    {
        const int ng = nbase + nl;
#pragma unroll
        for (int kc = 0; kc < 12; ++kc) {
            const int k0 = kc * 4 + hi * 2;
            bf[kc].x = __expf(trans[ng * T + k0]);
            bf[kc].y = __expf(trans[ng * T + k0 + 1]);
        }
    }

    // prologue: start streaming tile 0
    async_load_tile(feats + (size_t)b0 * T, lds0, tid);

    for (int t = 0; t < S; ++t) {
        __syncthreads();  // alpha writes + featS reads of step t-1 retired

        // issue async copy for step t+1 (double buffer)
        if (t + 1 < S)
            async_load_tile(feats + ((size_t)(t + 1) * B + b0) * T,
                            ((t + 1) & 1) ? lds1 : lds0, tid);

        // partial row-max over alpha (6 tag-groups of 8 per batch row)
        {
            const int b = tid & 15, g = tid >> 4;
            float mx = alphaS[b * 48 + g * 8];
#pragma unroll
            for (int j = 1; j < 8; ++j)
                mx = fmaxf(mx, alphaS[b * 48 + g * 8 + j]);
            redS[g * 16 + b] = mx;
        }
        __syncthreads();
        if (tid < 16) {
            float mx = redS[tid];
#pragma unroll
            for (int g = 1; g < 6; ++g) mx = fmaxf(mx, redS[g * 16 + tid]);
            mS[tid] = mx;
        }
        __syncthreads();
#pragma unroll
        for (int r = 0; r < 8; ++r) {
            const int idx = tid + r * 96;
            Ps[idx] = __expf(alphaS[idx] - mS[idx / 48]);
        }

        // ensure this wave's copy of tile t has landed (in-order ASYNCcnt:
        // tile t+1's 2 ops may remain outstanding), then barrier makes it
        // visible block-wide together with Ps.
        if (t + 1 < S) asm volatile("s_wait_asynccnt 0x2" ::: "memory");
        else           asm volatile("s_wait_asynccnt 0x0" ::: "memory");
        __syncthreads();

        // ---- WMMA phase: D(16x16) = P(16x48) @ E^T(48x16), K in chunks of 4.
        // A layout (16x4 f32): lane holds M = lane%16, K = 2*hi + c.
        v8f acc = {};
#pragma unroll
        for (int kc = 0; kc < 12; ++kc) {
            v2f a;
            const int ab = nl * 48 + kc * 4 + hi * 2;  // P[M][K], K,K+1 contiguous
            a.x = Ps[ab];
            a.y = Ps[ab + 1];
            acc = __builtin_amdgcn_wmma_f32_16x16x4_f32(
                false, a, false, bf[kc], (short)0, acc, false, false);
        }

        // ---- writeback: D layout: VGPR r -> row M = r + 8*hi, col N = lane%16
        const float* ft = featS[t & 1];
#pragma unroll
        for (int r = 0; r < 8; ++r) {
            const int mrow = r + 8 * hi;
            const int col  = nbase + nl;
            const float val = ft[mrow * 48 + col] + mS[mrow] + __logf(acc[r]);
            const float old = alphaS[mrow * 48 + col];
            alphaS[mrow * 48 + col] = (lensS[mrow] > t) ? val : old;
        }
    }

    __syncthreads();
    // final: fwd[b] = logsumexp_j(alpha[b,j] + trans[STOP][j])
    if (tid < 16) {
        const int b = tid;
        float mx = -3.0e38f;
        for (int j = 0; j < T; ++j)
            mx = fmaxf(mx, alphaS[b * 48 + j] + trans[STOP * T + j]);
        float s = 0.0f;
        for (int j = 0; j < T; ++j)
            s += __expf(alphaS[b * 48 + j] + trans[STOP * T + j] - mx);
        fwd[b0 + b] = mx + __logf(s);
    }
}

// ---------------------------------------------------------------------------
// Kernel 2: gold path score. One wave per batch element, lanes stride over S.
// ---------------------------------------------------------------------------
__global__ void crf_gold_kernel(const float* __restrict__ feats,
                                const float* __restrict__ trans,
                                const int* __restrict__ tags,
                                const int* __restrict__ lens,
                                float* __restrict__ gold) {
    const int tid  = threadIdx.x;
    const int lane = tid & 31;
    const int b    = blockIdx.x * (blockDim.x / 32) + (tid >> 5);
    if (b >= B) return;
    const int L = lens[b];

    float acc = 0.0f;
    // emission: sum_{s < L} feats[s, b, tags[b,s]]
    for (int s = lane; s < S; s += 32) {
        if (s < L) {
            const int tg = tags[b * S + s];
            acc += feats[((size_t)s * B + b) * T + tg];
        }
    }
    // transitions: sum_{p <= L} trans[pad(p+1)][pad(p)],
    // pad(0)=START, pad(p)=tags[p-1] if p-1<L else STOP (p in 1..S), pad(S+1)=STOP
    for (int p = lane; p <= S; p += 32) {
        if (p < L + 1) {
            const int prev = (p == 0) ? START
                                      : ((p - 1 < L) ? tags[b * S + p - 1] : STOP);
            const int next = (p + 1 <= S) ? ((p < L) ? tags[b * S + p] : STOP)
                                          : STOP;
            acc += trans[next * T + prev];
        }
    }
#pragma unroll
    for (int off = 16; off > 0; off >>= 1)
        acc += __shfl_down(acc, off, 32);
    if (lane == 0) gold[b] = acc;
}

// ---------------------------------------------------------------------------
// Kernel 3: deterministic mean of (fwd - gold).
// ---------------------------------------------------------------------------
__global__ void crf_reduce_kernel(const float* __restrict__ fwd,
                                  const float* __restrict__ gold,
                                  float* __restrict__ out) {
    __shared__ float sm[256];
    const int tid = threadIdx.x;
    float v = 0.0f;
    for (int i = tid; i < B; i += 256) v += fwd[i] - gold[i];
    sm[tid] = v;
    __syncthreads();
    for (int o = 128; o > 0; o >>= 1) {
        if (tid < o) sm[tid] += sm[tid + o];
        __syncthreads();
    }
    if (tid == 0) out[0] = sm[0] * (1.0f / (float)B);
}

extern "C" void kernel_launch(void* const* d_in, const int* in_sizes, int n_in,
                              void* d_out, int out_size, void* d_ws, size_t ws_size,
                              hipStream_t stream) {
    const float* feats = (const float*)d_in[0];
    const float* trans = (const float*)d_in[1];
    const int*   tags  = (const int*)d_in[2];
    const int*   lens  = (const int*)d_in[3];
    float* out = (float*)d_out;

    float* fwd  = (float*)d_ws;        // B floats
    float* gold = fwd + B;             // B floats

    crf_forward_kernel<<<B / 16, 96, 0, stream>>>(feats, trans, lens, fwd);
    crf_gold_kernel<<<B / 8, 256, 0, stream>>>(feats, trans, tags, lens, gold);
    crf_reduce_kernel<<<1, 256, 0, stream>>>(fwd, gold, out);
}